// TransformerBlock_66005057405491
// MI455X (gfx1250) — compile-verified
//
#include <hip/hip_runtime.h>
#include <math.h>

typedef __attribute__((ext_vector_type(2))) float v2f;
typedef __attribute__((ext_vector_type(8))) float v8f;

namespace {
constexpr int KW      = 5;                 // window
constexpr int PAD     = 2;                 // KW/2
constexpr int Cc      = 128;               // total channels
constexpr int Gg      = 4;                 // groups
constexpr int CG      = 32;                // channels per group
constexpr int Hh      = 96;
constexpr int Ww      = 96;
constexpr int TILE    = 8;                 // output tile (8x8 pixels per workgroup)
constexpr int PT      = TILE + 2 * PAD;    // 12 (padded tile edge)
constexpr int NPIX    = PT * PT;           // 144 padded pixels
constexpr int XSTR    = NPIX + 4;          // 148: xs stride; 8*148 % 64 == 32 -> conflict-free halves
constexpr int KVSTR   = NPIX + 2;          // 146 (float2 units); 8*146*2 % 64 == 32 -> conflict-free halves
constexpr int NCHUNK  = NPIX / 16;         // 9 pixel chunks of 16 for k/v WMMA
constexpr int THREADS = 128;               // 4 wave32
constexpr int NWAVES  = THREADS / 32;
constexpr float LOG2E = 1.4426950408889634f;
}

__device__ __forceinline__ int refl(int i, int n) {
  i = (i < 0) ? -i : i;
  return (i > n - 1) ? (2 * (n - 1) - i) : i;
}

// Fused: grouped 1x1 convs (q,k,v) via V_WMMA_F32_16X16X4_F32 + per-channel
// 5x5 window softmax-attention, all on-chip. k (and v) stored as channel-pair
// float2 {c, c+16} so the softmax runs on packed V_PK_*_F32 math with zero
// repacking; the conv phase writes the pair as two adjacent b32 components
// (fuses to DS_STORE_2ADDR_B32, no register repacking).
__global__ __launch_bounds__(THREADS)
void natt_fused_kernel(const float* __restrict__ x,
                       const float* __restrict__ wq,
                       const float* __restrict__ wk,
                       const float* __restrict__ wv,
                       float* __restrict__ out) {
  __shared__ float xs[CG * XSTR];          // input tile [ci][padded pix]
  __shared__ v2f   k2[(CG / 2) * KVSTR];   // key   [ch-pair][pix] = {k_m, k_m+16}
  __shared__ v2f   v2[(CG / 2) * KVSTR];   // value [ch-pair][pix] = {v_m, v_m+16}

  const int tid   = threadIdx.x;
  const int lane  = tid & 31;
  const int wv_id = tid >> 5;              // wave 0..3
  const int half  = lane >> 4;             // lane half (WMMA K / M split)
  const int lm    = lane & 15;

  const int tileX = blockIdx.x, tileY = blockIdx.y;
  const int b = blockIdx.z >> 2, g = blockIdx.z & 3;
  const int h0 = tileY * TILE, w0 = tileX * TILE;

  // ---- Stage x tile (reflect halo) into LDS, layout [ci][pix] ----
  const float* xg = x + (size_t)(b * Cc + g * CG) * (Hh * Ww);
  for (int t = tid; t < CG * NPIX; t += THREADS) {       // 36 iters, uniform
    int ch = t / NPIX;
    int p  = t - ch * NPIX;
    int py = p / PT, px = p - py * PT;
    int gh = refl(h0 + py - PAD, Hh);
    int gw = refl(w0 + px - PAD, Ww);
    xs[ch * XSTR + p] = xg[(size_t)ch * (Hh * Ww) + gh * Ww + gw];
  }
  __syncthreads();

  // ---- k and v grouped 1x1 conv over padded tile via WMMA ----
  // D(16co x 16pix) = A(16co x 4ci) * B(4ci x 16pix) + C, K-loop of 8 over ci=32.
  // Uniform trip count: 3 iterations/wave; chunk 8 recomputed redundantly
  // (same-value writes) so EXEC is provably all-ones around every WMMA.
  for (int cv = 0; cv < 2; ++cv) {
    const float* Wg = (cv == 0 ? wk : wv) + g * CG * CG; // [co][ci]
    float* dstf = (float*)((cv == 0) ? k2 : v2);
    v2f aw0[8], aw1[8];  // A fragments, co-blocks 0..15 and 16..31
#pragma unroll
    for (int kk = 0; kk < 8; ++kk) {
      int kb = kk * 4 + half * 2;  // A layout: v0={K0,K2}, v1={K1,K3} per lane half
      aw0[kk].x = Wg[lm * CG + kb];
      aw0[kk].y = Wg[lm * CG + kb + 1];
      aw1[kk].x = Wg[(lm + 16) * CG + kb];
      aw1[kk].y = Wg[(lm + 16) * CG + kb + 1];
    }
#pragma unroll
    for (int it = 0; it < 3; ++it) {
      int pc = wv_id + it * NWAVES;
      pc = (pc < NCHUNK) ? pc : (NCHUNK - 1);
      const int p = pc * 16 + lm;  // N = lane&15 for both halves (B/D layout)
      v8f acc0 = {}; v8f acc1 = {};
#pragma unroll
      for (int kk = 0; kk < 8; ++kk) {
        int kb = kk * 4 + half * 2;
        v2f bf;                    // B mirrors A: v0={K0,K2}, v1={K1,K3}
        bf.x = xs[kb * XSTR + p];
        bf.y = xs[kb * XSTR + XSTR + p];
        acc0 = __builtin_amdgcn_wmma_f32_16x16x4_f32(
            false, aw0[kk], false, bf, (short)0, acc0, false, false);
        acc1 = __builtin_amdgcn_wmma_f32_16x16x4_f32(
            false, aw1[kk], false, bf, (short)0, acc1, false, false);
      }
      const int m2 = half * 8;     // channel-pair row base held by this lane
#pragma unroll
      for (int r = 0; r < 8; ++r) {
        // two adjacent b32 components -> DS_STORE_2ADDR_B32, no repacking movs
        float* qp = dstf + ((m2 + r) * KVSTR + p) * 2;
        qp[0] = acc0[r];           // channel m
        qp[1] = acc1[r];           // channel m+16
      }
    }
  }
  __syncthreads();

  // ---- q conv for this wave's 16 inner pixels (kept in accumulators) ----
  const float* Wq = wq + g * CG * CG;
  v2f aq0[8], aq1[8];
#pragma unroll
  for (int kk = 0; kk < 8; ++kk) {
    int kb = kk * 4 + half * 2;
    aq0[kk].x = Wq[lm * CG + kb];
    aq0[kk].y = Wq[lm * CG + kb + 1];
    aq1[kk].x = Wq[(lm + 16) * CG + kb];
    aq1[kk].y = Wq[(lm + 16) * CG + kb + 1];
  }
  const int pt_  = wv_id * 16 + lm;           // tile pixel 0..63
  const int ty   = pt_ >> 3, tx = pt_ & 7;
  const int pidx = (ty + PAD) * PT + (tx + PAD);

  v8f c0 = {}; v8f c1 = {};
#pragma unroll
  for (int kk = 0; kk < 8; ++kk) {
    int kb = kk * 4 + half * 2;
    v2f bf;
    bf.x = xs[kb * XSTR + pidx];
    bf.y = xs[kb * XSTR + XSTR + pidx];
    c0 = __builtin_amdgcn_wmma_f32_16x16x4_f32(
        false, aq0[kk], false, bf, (short)0, c0, false, false);
    c1 = __builtin_amdgcn_wmma_f32_16x16x4_f32(
        false, aq1[kk], false, bf, (short)0, c1, false, false);
  }

  // ---- per-channel 5x5 window softmax on packed channel pairs ----
  // Lane holds pixel (ty,tx); pair r = channels {half*8+r, half*8+r+16},
  // exactly {c0[r], c1[r]}. One-pass softmax, exp(q*k)=exp2((q*log2e)*k),
  // packed V_PK_MUL/ADD/FMA_F32 throughout.
  const int gh = h0 + ty, gw = w0 + tx;
  const int wbase = ty * PT + tx;            // top-left tap of the window
  float* outp = out + ((size_t)(b * Cc + g * CG) * Hh + gh) * Ww + gw;
#pragma unroll
  for (int r = 0; r < 8; ++r) {
    const int m = half * 8 + r;
    v2f q2; q2.x = c0[r]; q2.y = c1[r];
    q2 *= LOG2E;
    const v2f* kr = k2 + m * KVSTR + wbase;
    const v2f* vr = v2 + m * KVSTR + wbase;
    v2f s2 = {}; v2f y2 = {};
#pragma unroll
    for (int di = 0; di < KW; ++di) {
#pragma unroll
      for (int dj = 0; dj < KW; ++dj) {
        const int idx = di * PT + dj;
        v2f kk = kr[idx];                      // ds_load_b64
        v2f vv = vr[idx];                      // ds_load_b64
        v2f l = q2 * kk;                       // v_pk_mul_f32
        v2f e;
        e.x = __builtin_amdgcn_exp2f(l.x);     // v_exp_f32
        e.y = __builtin_amdgcn_exp2f(l.y);     // v_exp_f32
        s2 += e;                               // v_pk_add_f32
        y2 = e * vv + y2;                      // v_pk_fma_f32 (contract)
      }
    }
    outp[(size_t)m * (Hh * Ww)]        = y2.x * __builtin_amdgcn_rcpf(s2.x);
    outp[(size_t)(m + 16) * (Hh * Ww)] = y2.y * __builtin_amdgcn_rcpf(s2.y);
  }
}

extern "C" void kernel_launch(void* const* d_in, const int* in_sizes, int n_in,
                              void* d_out, int out_size, void* d_ws, size_t ws_size,
                              hipStream_t stream) {
  (void)in_sizes; (void)n_in; (void)out_size; (void)d_ws; (void)ws_size;
  const float* x  = (const float*)d_in[0];
  const float* wq = (const float*)d_in[1];
  const float* wk = (const float*)d_in[2];
  const float* wv = (const float*)d_in[3];
  float* out = (float*)d_out;
  dim3 grid(Ww / TILE, Hh / TILE, 2 * Gg);   // (12, 12, 8)
  natt_fused_kernel<<<grid, THREADS, 0, stream>>>(x, wq, wk, wv, out);
}